// LogSumExp_43490838839590
// MI455X (gfx1250) — compile-verified
//
#include <hip/hip_runtime.h>
#include <stdint.h>

// ---------------------------------------------------------------------------
// log sigma_{4,5}(exp(x)) per row, x: (B, 8192) fp32, out: (2, B) fp32.
// out_j = j*M + log(e_j(exp(x - M))), truncated-poly recurrence, degree 5.
// CDNA5 path: TDM tensor_load_to_lds streams each 32KB row HBM->LDS,
// wave32 shuffle butterflies do the reductions.
// ---------------------------------------------------------------------------

namespace {
constexpr int kN       = 8192;            // n_classes (module constant)
constexpr int kThreads = 256;             // 8 wave32s
constexpr int kWaves   = kThreads / 32;
constexpr int kVecs    = kN / (kThreads * 4);  // 8 float4 loads per thread
}

typedef unsigned int u32x4 __attribute__((ext_vector_type(4)));
typedef int          i32x8 __attribute__((ext_vector_type(8)));
typedef int          i32x4 __attribute__((ext_vector_type(4)));

__global__ __launch_bounds__(kThreads)
void esp_log45_kernel(const float* __restrict__ x, float* __restrict__ out,
                      int nrows) {
    __shared__ float srow[kN];                  // 32 KB staging for one row
    __shared__ float wmax[kWaves];
    __shared__ float wpoly[kWaves][5];

    const int row  = blockIdx.x;
    const int tid  = threadIdx.x;
    const int wave = tid >> 5;

    // ---- TDM: DMA the whole row (32 KB) global -> LDS (wave 0 issues) ----
    if (wave == 0) {
        uint64_t gaddr = (uint64_t)(uintptr_t)(x + (size_t)row * kN);
        uint32_t laddr = (uint32_t)(uintptr_t)(&srow[0]);  // low 32 bits = LDS byte offset

        u32x4 g0;
        g0[0] = 1u;                                        // count=1 (valid user D#)
        g0[1] = laddr;                                     // lds_addr
        g0[2] = (uint32_t)(gaddr & 0xFFFFFFFFu);           // global_addr[31:0]
        g0[3] = (uint32_t)((gaddr >> 32) & 0x01FFFFFFu)    // global_addr[56:32]
              | 0x80000000u;                               // type=2 ("image")

        i32x8 g1;
        g1[0] = 0x00020000;                                // data_size=2 (4B), mask=0
        g1[1] = (int)(((uint32_t)kN & 0xFFFFu) << 16);     // tensor_dim0[15:0]
        g1[2] = 0x00010000;                                // tensor_dim0 hi=0 | tensor_dim1=1
        g1[3] = (int)(((uint32_t)kN & 0xFFFFu) << 16);     // tile_dim0 = 8192
        g1[4] = 1;                                         // tile_dim1=1, tile_dim2=0
        g1[5] = kN;                                        // tensor_dim0_stride
        g1[6] = 0;
        g1[7] = 0;

        i32x4 g2 = {0, 0, 0, 0};
        i32x4 g3 = {0, 0, 0, 0};
        i32x8 gx = {0, 0, 0, 0, 0, 0, 0, 0};               // extra operand (clang-23 form)

        __builtin_amdgcn_tensor_load_to_lds(g0, g1, g2, g3, gx, 0);
        __builtin_amdgcn_s_wait_tensorcnt(0);
    }
    __syncthreads();

    // ---- LDS -> registers (8 x b128 per thread), local max ----
    float4 v[kVecs];
    float m = -INFINITY;
#pragma unroll
    for (int i = 0; i < kVecs; ++i) {
        v[i] = *reinterpret_cast<const float4*>(&srow[i * (kThreads * 4) + tid * 4]);
        m = fmaxf(m, fmaxf(fmaxf(v[i].x, v[i].y), fmaxf(v[i].z, v[i].w)));
    }

    // wave32 max butterfly
#pragma unroll
    for (int off = 16; off >= 1; off >>= 1)
        m = fmaxf(m, __shfl_xor(m, off, 32));
    if ((tid & 31) == 0) wmax[wave] = m;
    __syncthreads();

    float M = wmax[0];
#pragma unroll
    for (int w = 1; w < kWaves; ++w) M = fmaxf(M, wmax[w]);

    // ---- per-thread truncated polynomial prod (1 + z_i t), degree <= 5 ----
    float c1 = 0.f, c2 = 0.f, c3 = 0.f, c4 = 0.f, c5 = 0.f;
#pragma unroll
    for (int i = 0; i < kVecs; ++i) {
        float zv[4] = {v[i].x, v[i].y, v[i].z, v[i].w};
#pragma unroll
        for (int j = 0; j < 4; ++j) {
            float z = __expf(zv[j] - M);  // z in (0,1]
            c5 = fmaf(z, c4, c5);
            c4 = fmaf(z, c3, c4);
            c3 = fmaf(z, c2, c3);
            c2 = fmaf(z, c1, c2);
            c1 += z;
        }
    }

    // ---- wave32 butterfly of truncated poly products (c0 == 1 both sides) ----
#pragma unroll
    for (int off = 1; off < 32; off <<= 1) {
        float d1 = __shfl_xor(c1, off, 32);
        float d2 = __shfl_xor(c2, off, 32);
        float d3 = __shfl_xor(c3, off, 32);
        float d4 = __shfl_xor(c4, off, 32);
        float d5 = __shfl_xor(c5, off, 32);
        float n1 = c1 + d1;
        float n2 = c2 + d2 + c1 * d1;
        float n3 = c3 + d3 + c1 * d2 + c2 * d1;
        float n4 = c4 + d4 + c1 * d3 + c2 * d2 + c3 * d1;
        float n5 = c5 + d5 + c1 * d4 + c2 * d3 + c3 * d2 + c4 * d1;
        c1 = n1; c2 = n2; c3 = n3; c4 = n4; c5 = n5;
    }

    if ((tid & 31) == 0) {
        wpoly[wave][0] = c1; wpoly[wave][1] = c2; wpoly[wave][2] = c3;
        wpoly[wave][3] = c4; wpoly[wave][4] = c5;
    }
    __syncthreads();

    // ---- fold the 8 wave polynomials, emit the two outputs ----
    if (tid == 0) {
        float C1 = wpoly[0][0], C2 = wpoly[0][1], C3 = wpoly[0][2],
              C4 = wpoly[0][3], C5 = wpoly[0][4];
#pragma unroll
        for (int w = 1; w < kWaves; ++w) {
            float d1 = wpoly[w][0], d2 = wpoly[w][1], d3 = wpoly[w][2],
                  d4 = wpoly[w][3], d5 = wpoly[w][4];
            float n1 = C1 + d1;
            float n2 = C2 + d2 + C1 * d1;
            float n3 = C3 + d3 + C1 * d2 + C2 * d1;
            float n4 = C4 + d4 + C1 * d3 + C2 * d2 + C3 * d1;
            float n5 = C5 + d5 + C1 * d4 + C2 * d3 + C3 * d2 + C4 * d1;
            C1 = n1; C2 = n2; C3 = n3; C4 = n4; C5 = n5;
        }
        out[row]         = 4.0f * M + logf(C4);   // log e_4(exp x)
        out[nrows + row] = 5.0f * M + logf(C5);   // log e_5(exp x)
    }
}

extern "C" void kernel_launch(void* const* d_in, const int* in_sizes, int n_in,
                              void* d_out, int out_size, void* d_ws, size_t ws_size,
                              hipStream_t stream) {
    const float* x = (const float*)d_in[0];
    float* out = (float*)d_out;
    const int nrows = in_sizes[0] / kN;   // B = 2048 for the reference shapes
    hipLaunchKernelGGL(esp_log45_kernel, dim3(nrows), dim3(kThreads), 0, stream,
                       x, out, nrows);
}